// SIM_Model_8349416423957
// MI455X (gfx1250) — compile-verified
//
#include <hip/hip_runtime.h>

// 4-wide float vector that lowers to b128 global memory ops.
typedef float v4f __attribute__((ext_vector_type(4)));

#define SIM_STEPS 100
#define SIM_G     20.0f

// One thread advances 4 independent recurrences fully in registers and
// streams y/h history with non-temporal 128-bit stores (write-once data,
// 200MB > 192MB L2, so bypass with TH_STORE_NT).
__global__ __launch_bounds__(256) void sim_scan_kernel(
    const float* __restrict__ h0,
    const float* __restrict__ alpha1_p,
    const float* __restrict__ alpha2_p,
    const float* __restrict__ theta_p,
    float* __restrict__ out,
    int n)
{
    // Wave-uniform scalar parameter loads -> s_load_b32.
    const float a1 = alpha1_p[0];
    const float a2 = alpha2_p[0];
    const float th = theta_p[0];
    // Loop-invariant: denom = 1 - a1*(1-th); hoist the divide out of the loop.
    const float inv_denom = 1.0f / (1.0f - a1 * (1.0f - th));

    float* __restrict__ y_out = out;                                   // [STEPS, n]
    float* __restrict__ h_out = out + (size_t)SIM_STEPS * (size_t)n;   // [STEPS, n]

    const int nv = n >> 2;  // number of full float4 groups
    const int t  = blockIdx.x * blockDim.x + threadIdx.x;

    if (t < nv) {
        // Stream-in the initial state (read-once -> non-temporal).
        v4f h = __builtin_nontemporal_load((const v4f*)h0 + t);
        const size_t base = (size_t)t * 4;

        #pragma unroll 4
        for (int s = 0; s < SIM_STEPS; ++s) {
            // y  = (G + a2*h) / denom
            // yd = y - th*y
            // c  = a1*yd + a2*h
            // h  = h + (yd - c)
            v4f y  = (SIM_G + a2 * h) * inv_denom;
            v4f yd = y - th * y;
            v4f c  = a1 * yd + a2 * h;
            h      = h + (yd - c);

            const size_t off = (size_t)s * (size_t)n + base;
            __builtin_nontemporal_store(y, (v4f*)(y_out + off));  // global_store_b128 th:TH_STORE_NT
            __builtin_nontemporal_store(h, (v4f*)(h_out + off));
        }
    } else if (t == nv) {
        // Scalar tail (n % 4 elements); no-op when n is a multiple of 4.
        const int rem = n - (nv << 2);
        for (int r = 0; r < rem; ++r) {
            const int i = (nv << 2) + r;
            float h = h0[i];
            for (int s = 0; s < SIM_STEPS; ++s) {
                float y  = (SIM_G + a2 * h) * inv_denom;
                float yd = y - th * y;
                float c  = a1 * yd + a2 * h;
                h        = h + (yd - c);
                const size_t off = (size_t)s * (size_t)n + (size_t)i;
                __builtin_nontemporal_store(y, y_out + off);
                __builtin_nontemporal_store(h, h_out + off);
            }
        }
    }
}

extern "C" void kernel_launch(void* const* d_in, const int* in_sizes, int n_in,
                              void* d_out, int out_size, void* d_ws, size_t ws_size,
                              hipStream_t stream) {
    (void)n_in; (void)out_size; (void)d_ws; (void)ws_size;

    const float* h0 = (const float*)d_in[0];
    const float* a1 = (const float*)d_in[1];
    const float* a2 = (const float*)d_in[2];
    const float* th = (const float*)d_in[3];
    float*       out = (float*)d_out;
    const int    n   = in_sizes[0];

    // n/4 vector threads + 1 tail thread.
    const int nthreads = (n >> 2) + 1;
    const int block    = 256;                    // 8 wave32 waves per block
    const int grid     = (nthreads + block - 1) / block;

    sim_scan_kernel<<<grid, block, 0, stream>>>(h0, a1, a2, th, out, n);
}